// BiRNNDecoder_69106023792741
// MI455X (gfx1250) — compile-verified
//
#include <hip/hip_runtime.h>
#include <hip/hip_bf16.h>

#define HID    24
#define TSTEPS 262
#define KINFO  256

typedef __attribute__((ext_vector_type(2))) float v2f;
typedef __attribute__((ext_vector_type(8))) float v8f;

// One workgroup = 64 threads = 2 waves handling the SAME 16 batch rows:
//   wave 0: forward GRU, wave 1: backward GRU.
// Per wave, per time step: (16x28) @ (28x96) via 6 N-tiles x 7 K-chunks of
// V_WMMA_F32_16X16X4_F32, plus 2 WMMAs for the separate n-gate input term.
// A = [h(24) | x0 x1 1 0]; B = combined [W_hh ; W_ih ; b_ih+b_hh] staged in LDS.
__global__ __launch_bounds__(64) void birnn_gru_wmma(
    const float* __restrict__ x,
    const float* __restrict__ w_ih_f, const float* __restrict__ w_hh_f,
    const float* __restrict__ b_ih_f, const float* __restrict__ b_hh_f,
    const float* __restrict__ w_ih_b, const float* __restrict__ w_hh_b,
    const float* __restrict__ b_ih_b, const float* __restrict__ b_hh_b,
    const float* __restrict__ w_out,  const float* __restrict__ b_out,
    float* __restrict__ out)
{
    __shared__ float Wlds[2][28][96];       // logical combined weights per dir
    __shared__ float Wgi [2][4][32];        // n-gate input-projection weights
    __shared__ float hls [2][16][32];       // per-wave h transpose buffer
    __shared__ float lbuf[2][16][KINFO + 1];// per-dir logits (padded stride)

    const int tid  = threadIdx.x;
    const int base = blockIdx.x * 16;       // batch tile base

    // ---- stage logical weight matrices into LDS (one-time) ----
    for (int idx = tid; idx < 2 * 28 * 96; idx += 64) {
        int d = idx / (28 * 96);
        int rem = idx % (28 * 96);
        int k = rem / 96, n = rem % 96;
        int g = n >> 5, c = n & 31;         // gate (r,z,n) and within-gate col
        const float* wih = d ? w_ih_b : w_ih_f;
        const float* whh = d ? w_hh_b : w_hh_f;
        const float* bih = d ? b_ih_b : b_ih_f;
        const float* bhh = d ? b_hh_b : b_hh_f;
        float v = 0.0f;
        if (c < HID) {
            int row = g * HID + c;          // PyTorch gate-row order [r,z,n]
            if      (k < HID)  v = whh[row * HID + k];
            else if (k == 24)  v = (g < 2) ? wih[row * 2 + 0] : 0.0f;
            else if (k == 25)  v = (g < 2) ? wih[row * 2 + 1] : 0.0f;
            else if (k == 26)  v = (g < 2) ? (bih[row] + bhh[row]) : bhh[row];
        }
        Wlds[d][k][n] = v;
    }
    for (int idx = tid; idx < 2 * 4 * 32; idx += 64) {
        int d = idx / 128; int rem = idx % 128;
        int kk = rem / 32, c = rem % 32;
        const float* wih = d ? w_ih_b : w_ih_f;
        const float* bih = d ? b_ih_b : b_ih_f;
        float v = 0.0f;
        if (c < HID) {
            int row = 2 * HID + c;          // n-gate rows 48..71
            if      (kk == 0) v = wih[row * 2 + 0];
            else if (kk == 1) v = wih[row * 2 + 1];
            else if (kk == 2) v = bih[row];
        }
        Wgi[d][kk][c] = v;
    }
    __syncthreads();

    const int wave = tid >> 5;              // 0 = forward, 1 = backward
    const int lane = tid & 31;
    const int col  = lane & 15;
    const int lh   = lane >> 4;             // lane half
    const int kp   = lh * 2;                // K offset within 4-chunk
    const int roff = lh * 8;                // row offset in D layout

    // ---- load B fragments into registers ----
    // 32-bit B frag (4x16): VGPR0 = row K0 (lanes0-15) / K2 (lanes16-31),
    //                       VGPR1 = row K1 / K3.
    v2f Bf[6][7];
    #pragma unroll
    for (int tt = 0; tt < 6; ++tt)
        #pragma unroll
        for (int ck = 0; ck < 7; ++ck) {
            Bf[tt][ck].x = Wlds[wave][4 * ck + kp    ][16 * tt + col];
            Bf[tt][ck].y = Wlds[wave][4 * ck + kp + 1][16 * tt + col];
        }
    v2f Bg[2];
    #pragma unroll
    for (int tt = 0; tt < 2; ++tt) {
        Bg[tt].x = Wgi[wave][kp    ][16 * tt + col];
        Bg[tt].y = Wgi[wave][kp + 1][16 * tt + col];
    }
    // per-lane output-projection weights matching A-fragment layout
    v2f woa[6];
    #pragma unroll
    for (int c = 0; c < 6; ++c) {
        woa[c].x = w_out[wave * HID + 4 * c + kp];
        woa[c].y = w_out[wave * HID + 4 * c + kp + 1];
    }

    float* hb = &hls[wave][0][0];

    // hidden state: hD = D layout (2 gate-tiles x 8 VGPRs), afr = A fragments
    float hD[2][8];
    v2f afr[7];
    #pragma unroll
    for (int i = 0; i < 2; ++i)
        #pragma unroll
        for (int j = 0; j < 8; ++j) hD[i][j] = 0.0f;
    #pragma unroll
    for (int c = 0; c < 6; ++c) { afr[c].x = 0.0f; afr[c].y = 0.0f; }

    const int nsteps = wave ? TSTEPS : KINFO;   // fwd only needs t<256
    const int xrow = (base + col) * TSTEPS;

    for (int s = 0; s < nsteps; ++s) {
        const int t = wave ? (TSTEPS - 1 - s) : s;

        // A chunk 6 = [x0, x1, 1, 0]
        const float x0 = x[(xrow + t) * 2 + 0];
        const float x1 = x[(xrow + t) * 2 + 1];
        afr[6].x = lh ? 1.0f : x0;
        afr[6].y = lh ? 0.0f : x1;

        // ---- matrix stage: gh/gi combined gates ----
        v8f acc[6], gacc[2];
        #pragma unroll
        for (int tt = 0; tt < 6; ++tt) {
            v8f c = {};
            #pragma unroll
            for (int ck = 0; ck < 7; ++ck)
                c = __builtin_amdgcn_wmma_f32_16x16x4_f32(
                        false, afr[ck], false, Bf[tt][ck],
                        (short)0, c, false, false);
            acc[tt] = c;
        }
        #pragma unroll
        for (int tt = 0; tt < 2; ++tt) {
            v8f z = {};
            gacc[tt] = __builtin_amdgcn_wmma_f32_16x16x4_f32(
                           false, afr[6], false, Bg[tt],
                           (short)0, z, false, false);
        }

        // ---- gate nonlinearities + state update (lane-aligned) ----
        #pragma unroll
        for (int i = 0; i < 2; ++i)
            #pragma unroll
            for (int j = 0; j < 8; ++j) {
                float r   = 1.0f / (1.0f + __expf(-acc[i][j]));
                float zz  = 1.0f / (1.0f + __expf(-acc[2 + i][j]));
                float pre = gacc[i][j] + r * acc[4 + i][j];
                float n   = 1.0f - 2.0f / (1.0f + __expf(2.0f * pre)); // tanh
                float hn  = n + zz * (hD[i][j] - n);
                hD[i][j]  = hn;
                hb[(j + roff) * 32 + 16 * i + col] = hn;   // D-layout -> LDS
            }

        // ---- transpose reload: next-step A fragments ----
        #pragma unroll
        for (int c = 0; c < 6; ++c) {
            afr[c].x = hb[col * 32 + 4 * c + kp];
            afr[c].y = hb[col * 32 + 4 * c + kp + 1];
        }

        // ---- fused output projection for this time step ----
        if (t < KINFO) {
            float p = 0.0f;
            #pragma unroll
            for (int c = 0; c < 6; ++c)
                p += afr[c].x * woa[c].x + afr[c].y * woa[c].y;
            p += __shfl_xor(p, 16, 32);     // combine the two lane halves
            if (lane < 16) lbuf[wave][lane][t] = p;
        }
    }

    __syncthreads();

    // ---- combine fwd + bwd + bias, coalesced store ----
    const float bo = b_out[0];
    for (int idx = tid; idx < 16 * KINFO; idx += 64) {
        int row = idx >> 8;
        int t   = idx & (KINFO - 1);
        out[(base + row) * KINFO + t] = lbuf[0][row][t] + lbuf[1][row][t] + bo;
    }
}

extern "C" void kernel_launch(void* const* d_in, const int* in_sizes, int n_in,
                              void* d_out, int out_size, void* d_ws, size_t ws_size,
                              hipStream_t stream) {
    const float* x      = (const float*)d_in[0];
    const float* w_ih_f = (const float*)d_in[1];
    const float* w_hh_f = (const float*)d_in[2];
    const float* b_ih_f = (const float*)d_in[3];
    const float* b_hh_f = (const float*)d_in[4];
    const float* w_ih_b = (const float*)d_in[5];
    const float* w_hh_b = (const float*)d_in[6];
    const float* b_ih_b = (const float*)d_in[7];
    const float* b_hh_b = (const float*)d_in[8];
    const float* w_out  = (const float*)d_in[9];
    const float* b_out  = (const float*)d_in[10];
    float* out = (float*)d_out;

    const int batch = 8192;
    dim3 grid(batch / 16, 1, 1);   // 512 workgroups x 2 waves = 1024 waves
    dim3 block(64, 1, 1);
    hipLaunchKernelGGL(birnn_gru_wmma, grid, block, 0, stream,
                       x, w_ih_f, w_hh_f, b_ih_f, b_hh_f,
                       w_ih_b, w_hh_b, b_ih_b, b_hh_b, w_out, b_out, out);
}